// Raycaster_79508434584085
// MI455X (gfx1250) — compile-verified
//
#include <hip/hip_runtime.h>
#include <math.h>

#define RES_W 256
#define RES_H 256
#define MAX_SAMPLES 384
#define TF_RES 128
#define NEAR_T 0.1f
#define FAR_T 100.0f
#define HIT_EPS 1e-3f
// tan(15 degrees)
#define TANF 0.26794919243112270647f

typedef __attribute__((ext_vector_type(4))) unsigned int u32x4;
typedef __attribute__((ext_vector_type(8))) int i32x8;
typedef __attribute__((ext_vector_type(4))) int i32x4;

__global__ __launch_bounds__(256) void Raycaster_kernel(
    const float* __restrict__ vol,   // [256,256,256]
    const float* __restrict__ tf,    // [128,4]
    const float* __restrict__ cam_p, // [3]
    const int*   __restrict__ sr_p,  // [1]
    float*       __restrict__ out)   // [256,256,5]
{
    __shared__ float4 sTF[TF_RES]; // 2 KB transfer function

    // ---- Stage TF table into LDS via the Tensor Data Mover (wave 0 only) ----
    if (threadIdx.x < 32) {
        unsigned lds_off = (unsigned)(unsigned long long)(&sTF[0]); // low 32 bits = LDS offset
        unsigned long long ga = (unsigned long long)tf;
        u32x4 g0;
        g0[0] = 1u;                                   // count=1, user descriptor
        g0[1] = lds_off;                              // lds_addr
        g0[2] = (unsigned)(ga & 0xFFFFFFFFull);       // global_addr[31:0]
        g0[3] = (unsigned)((ga >> 32) & 0x1FFFFFFull) // global_addr[56:32]
              | (2u << 30);                           // type = 2 ("image")
        i32x8 g1;
        g1[0] = (2 << 16);              // workgroup_mask=0, data_size=2 (4 bytes)
        g1[1] = (int)(512u << 16);      // tensor_dim0[15:0] = 512 (128 rgba4 rows)
        g1[2] = (int)(1u << 16);        // tensor_dim0 hi = 0, tensor_dim1 lo = 1
        g1[3] = (int)(512u << 16);      // tensor_dim1 hi = 0, tile_dim0 = 512
        g1[4] = 1;                      // tile_dim1 = 1, tile_dim2 = 0
        g1[5] = 512;                    // tensor_dim0_stride low32
        g1[6] = 0;                      // stride0 hi / stride1 lo
        g1[7] = 0;                      // stride1 hi
        i32x4 gz4 = {0, 0, 0, 0};                     // groups 2/3 unused (<=2D tensor)
        i32x8 gz8 = {0, 0, 0, 0, 0, 0, 0, 0};         // extra group (6-arg toolchain form)
        __builtin_amdgcn_tensor_load_to_lds(g0, g1, gz4, gz4, gz8, 0);
        __builtin_amdgcn_s_wait_tensorcnt(0);
    }
    __syncthreads();

    // ---- Pixel assignment: 16x16 tile per block, 16x2 footprint per wave ----
    const int tid = threadIdx.x;
    const int w = ((int)blockIdx.x & 15) * 16 + (tid & 15);
    const int h = ((int)blockIdx.x >> 4) * 16 + (tid >> 4);

    // ---- Camera basis ----
    const float cx = cam_p[0], cy = cam_p[1], cz = cam_p[2];
    float n = sqrtf(cx * cx + cy * cy + cz * cz);
    const float fx = -cx / n, fy = -cy / n, fz = -cz / n;
    // right = normalize(cross(fwd,(0,1,0))) = normalize((-fz,0,fx))
    float rx = -fz, rz = fx;
    n = sqrtf(rx * rx + rz * rz);
    rx /= n; rz /= n;
    // up = cross(right, fwd); right.y == 0
    const float ux = -rz * fy;
    const float uy = rz * fx - rx * fz;
    const float uz = rx * fy;

    const float uu = (((float)w + 0.5f) * (2.0f / RES_W) - 1.0f) * TANF; // aspect = 1
    const float vv = (((float)h + 0.5f) * (2.0f / RES_H) - 1.0f) * TANF;
    float dx = fx + uu * rx + vv * ux;
    float dy = fy + vv * uy;               // right.y == 0
    float dz = fz + uu * rz + vv * uz;
    n = sqrtf(dx * dx + dy * dy + dz * dz);
    dx /= n; dy /= n; dz /= n;

    // ---- Ray / [-1,1]^3 box intersection ----
    const float ivx = 1.0f / dx, ivy = 1.0f / dy, ivz = 1.0f / dz;
    const float t0x = (-1.0f - cx) * ivx, t1x = (1.0f - cx) * ivx;
    const float t0y = (-1.0f - cy) * ivy, t1y = (1.0f - cy) * ivy;
    const float t0z = (-1.0f - cz) * ivz, t1z = (1.0f - cz) * ivz;
    const float tmin = fmaxf(fmaxf(fminf(t0x, t1x), fminf(t0y, t1y)), fminf(t0z, t1z));
    const float tmax = fminf(fminf(fmaxf(t0x, t1x), fmaxf(t0y, t1y)), fmaxf(t0z, t1z));
    const bool hit = (tmax >= 0.0f) && (tmin <= tmax);
    const float entry = fmaxf(tmin, NEAR_T);
    const float dist = hit ? fmaxf(tmax - entry, 0.0f) : 0.0f;

    const float sr = (float)sr_p[0];
    const float inv_sr = 1.0f / sr;
    const bool sr_is_one = (inv_sr == 1.0f);
    const float ns = fminf(fmaxf(ceilf(dist * 128.0f * sr), 1.0f), (float)MAX_SAMPLES);
    const float step = dist / ns;
    const int nsi = (int)ns;

    // ---- March ----
    float Tr = 1.0f, r = 0.0f, g = 0.0f, b = 0.0f, acc = 0.0f, depth = 1.0f;
    bool found = false;

    if (hit) {
        for (int s = 0; s < nsi; ++s) {
            const float t = entry + ((float)s + 0.5f) * step;
            const float px = fmaf(t, dx, cx);
            const float py = fmaf(t, dy, cy);
            const float pz = fmaf(t, dz, cz);
            // voxel coords, clamped to [0, 255] (fp32(255 - 1e-6) == 255.0f)
            const float txc = fminf(fmaxf(fmaf(px, 127.5f, 127.5f), 0.0f), 255.0f);
            const float tyc = fminf(fmaxf(fmaf(py, 127.5f, 127.5f), 0.0f), 255.0f);
            const float tzc = fminf(fmaxf(fmaf(pz, 127.5f, 127.5f), 0.0f), 255.0f);
            const float flx = floorf(txc), fly = floorf(tyc), flz = floorf(tzc);
            const float ffx = txc - flx, ffy = tyc - fly, ffz = tzc - flz;
            const int x0 = (int)flx, y0 = (int)fly, z0 = (int)flz;
            const int x1 = min(x0 + 1, 255);
            const int y1 = min(y0 + 1, 255);
            const int z1 = min(z0 + 1, 255);
            const int ix0 = x0 << 16, ix1 = x1 << 16;
            const int iy0 = y0 << 8,  iy1 = y1 << 8;
            // 8 gathers (addresses computed up front -> batched issue)
            const float c000 = vol[ix0 + iy0 + z0];
            const float c100 = vol[ix1 + iy0 + z0];
            const float c010 = vol[ix0 + iy1 + z0];
            const float c110 = vol[ix1 + iy1 + z0];
            const float c001 = vol[ix0 + iy0 + z1];
            const float c101 = vol[ix1 + iy0 + z1];
            const float c011 = vol[ix0 + iy1 + z1];
            const float c111 = vol[ix1 + iy1 + z1];
            const float omx = 1.0f - ffx;
            const float c00 = c000 * omx + c100 * ffx;
            const float c10 = c010 * omx + c110 * ffx;
            const float c01 = c001 * omx + c101 * ffx;
            const float c11 = c011 * omx + c111 * ffx;
            const float omy = 1.0f - ffy;
            const float c0 = c00 * omy + c10 * ffy;
            const float c1 = c01 * omy + c11 * ffy;
            const float I = c0 * (1.0f - ffz) + c1 * ffz;

            // transfer function lookup (LDS, float4)
            const float xi = fminf(fmaxf(I, 0.0f), 1.0f) * (float)(TF_RES - 1);
            const float fll = floorf(xi);
            const int l0 = (int)fll;
            const int l1 = min(l0 + 1, TF_RES - 1);
            const float tfr = xi - fll;
            const float4 e0 = sTF[l0];
            const float4 e1 = sTF[l1];
            const float aR = e0.w + tfr * (e1.w - e0.w);

            // opacity correction for sampling rate
            const float a = sr_is_one ? aR : (1.0f - __powf(1.0f - aR, inv_sr));

            if (!found && a > HIT_EPS) {
                found = true;
                depth = (t - NEAR_T) * (1.0f / (FAR_T - NEAR_T));
            }
            const float wgt = Tr * a;
            r = fmaf(wgt, e0.x + tfr * (e1.x - e0.x), r);
            g = fmaf(wgt, e0.y + tfr * (e1.y - e0.y), g);
            b = fmaf(wgt, e0.z + tfr * (e1.z - e0.z), b);
            acc += wgt;
            Tr *= (1.0f - a);
            if (Tr < 1e-5f) break;   // early ray termination (below fp32 noise)
        }
    }

    // BG == 0, so rgb_out == rgb
    float* o = out + ((size_t)w * RES_H + h) * 5;
    o[0] = r; o[1] = g; o[2] = b; o[3] = acc; o[4] = depth;
}

extern "C" void kernel_launch(void* const* d_in, const int* in_sizes, int n_in,
                              void* d_out, int out_size, void* d_ws, size_t ws_size,
                              hipStream_t stream) {
    const float* vol = (const float*)d_in[0];   // [1,256,256,256] f32
    const float* tf  = (const float*)d_in[1];   // [1,128,4] f32
    const float* cam = (const float*)d_in[2];   // [1,3] f32
    const int*   sr  = (const int*)d_in[3];     // scalar int
    float* out = (float*)d_out;                 // [1,256,256,5] f32
    (void)in_sizes; (void)n_in; (void)out_size; (void)d_ws; (void)ws_size;

    dim3 grid(256);   // 16x16 tiles of 16x16 pixels
    dim3 block(256);  // 8 wave32s
    Raycaster_kernel<<<grid, block, 0, stream>>>(vol, tf, cam, sr, out);
}